// DWT_10187662426901
// MI455X (gfx1250) — compile-verified
//
#include <hip/hip_runtime.h>

typedef float v2f __attribute__((ext_vector_type(2)));
typedef float v8f __attribute__((ext_vector_type(8)));

// bior3.5 analysis filters (pywt conventions), dec_len = 12.
// dec_lo = sqrt(2)/2048 * [-20,60,76,-388,-104,1400,1400,-104,-388,76,60,-20]
// dec_hi = sqrt(2)/8   * [0,0,0,0,-1,3,-3,1,0,0,0,0]
__constant__ float c_dec[2][12] = {
    {-0.013810679320049756f,  0.041432037960149270f,  0.052480581416221075f,
     -0.267927178088965270f, -0.071815532464258730f,  0.966747552403482900f,
      0.966747552403482900f, -0.071815532464258730f, -0.267927178088965270f,
      0.052480581416221075f,  0.041432037960149270f, -0.013810679320049756f},
    { 0.0f, 0.0f, 0.0f, 0.0f,
     -0.176776695296636900f,  0.530330085889910700f,
     -0.530330085889910700f,  0.176776695296636900f,
      0.0f, 0.0f, 0.0f, 0.0f}
};

#define TS       16   // WMMA tile edge
#define OT       32   // output tile edge per workgroup (per subband)
#define WIN      80   // shared input window edge (2*OT + 10 halo -> pad to 80)
#define KC       48   // padded inner contraction per 16-output tile (12 chunks of 4)
#define WSTRIDE  82   // LDS window row stride (floats); 82*d % 64 != 0 for d in 1..31
#define ISTRIDE  33   // intermediate row stride (odd -> conflict free)
#define FSTRIDE  49   // filter table row stride (48 cols + 1 pad)

// 4 waves (128 threads) per workgroup compute a 32x32 output tile for all 4
// subbands of one image at one level. Fused row+column transform through LDS;
// all matrix work uses V_WMMA_F32_16X16X4_F32 (full fp32 precision).
// Halo amplification: (80/64)^2 = 1.56x vs 2.25x for 16x16 tiles.
__global__ __launch_bounds__(128) void dwt2_level_kernel(
    const float* __restrict__ in,   // (24, H, W)
    float* __restrict__ aa,         // (24, H/2, W/2)
    float* __restrict__ hi_base,    // da slab; ad = +slab, dd = +2*slab
    size_t slab,                    // elements per subband slab = 24*(H/2)*(W/2)
    int H, int W)
{
    const int tid  = threadIdx.x;
    const int lane = tid & 31;
    const int wave = tid >> 5;              // 0..3
    const int img  = blockIdx.z;
    const int ox0  = blockIdx.x * OT;       // output tile origin (x)
    const int oy0  = blockIdx.y * OT;       // output tile origin (y)
    const int h = H >> 1, w = W >> 1;

    __shared__ float s_win[WIN][WSTRIDE];       // 80x80 input window
    __shared__ float s_int[2][WIN][ISTRIDE];    // row-pass lo/hi: [f][row][t 0..31]
    __shared__ float s_flt[2][TS][FSTRIDE];     // Mt[f][t][c] = dec_f[2t+13-c]

    // Build banded filter matrices (shared by every tile-matmul in both passes).
    for (int idx = tid; idx < 2 * TS * KC; idx += 128) {
        int f = idx / (TS * KC);
        int r = idx % (TS * KC);
        int t = r / KC;
        int c = r % KC;
        int j = 2 * t + 13 - c;
        s_flt[f][t][c] = (j >= 0 && j < 12) ? c_dec[f][j] : 0.0f;
    }

    // Load periodized 80x80 window (dims are powers of two -> mask wrap).
    const float* src = in + (size_t)img * H * W;
    const int gx0 = 2 * ox0 - 12;
    const int gy0 = 2 * oy0 - 12;
    const unsigned mw = (unsigned)(W - 1), mh = (unsigned)(H - 1);
    for (int idx = tid; idx < WIN * WIN; idx += 128) {
        int wy = idx / WIN, wx = idx % WIN;
        unsigned gy = (unsigned)(gy0 + wy) & mh;
        unsigned gx = (unsigned)(gx0 + wx) & mw;
        s_win[wy][wx] = src[(size_t)gy * W + gx];
    }
    __syncthreads();

    const int mrow  = lane & 15;            // A: M index / B,D: N index
    const int khalf = (lane >> 4) << 1;     // K sub-index base: 0 or 2
    const int dhalf = (lane >> 4) << 3;     // D: +8 rows for upper half-wave

    // ---------------- Row transform (along x) ----------------
    // s_int[f][row][q*16+t] = sum_c Mt[f][t][c] * s_win[row][32q + c]
    // 20 tile-matmuls (5 rowgroups x 2 col-groups x 2 filters), 5 per wave.
#pragma unroll
    for (int i = 0; i < 5; ++i) {
        const int tt = wave + 4 * i;        // 0..19
        const int f  = tt & 1;
        const int q  = (tt >> 1) & 1;
        const int g  = tt >> 2;             // 0..4
        v8f acc = {0.f, 0.f, 0.f, 0.f, 0.f, 0.f, 0.f, 0.f};
#pragma unroll
        for (int k = 0; k < 12; ++k) {
            const int cf = 4 * k + khalf;       // filter-table column
            const int cw = 32 * q + cf;         // window column
            v2f a;  // A(16x4): data rows
            a.x = s_win[g * 16 + mrow][cw];
            a.y = s_win[g * 16 + mrow][cw + 1];
            v2f b;  // B(4x16): filter matrix
            b.x = s_flt[f][mrow][cf];
            b.y = s_flt[f][mrow][cf + 1];
            acc = __builtin_amdgcn_wmma_f32_16x16x4_f32(
                false, a, false, b, (short)0, acc, false, false);
        }
        const int rbase = g * 16 + dhalf;
#pragma unroll
        for (int v = 0; v < 8; ++v)
            s_int[f][rbase + v][q * 16 + mrow] = acc[v];
    }
    __syncthreads();

    // ---------------- Column transform (along y) + store ----------------
    // out[fy][fx][ry*16+t][qx*16+col] = sum_c Mt[fy][t][c] * s_int[fx][32ry + c][qx*16+col]
    // 16 tile-matmuls (fy x fx x 2 row-quadrants x 2 col-quadrants), 4 per wave.
#pragma unroll
    for (int i = 0; i < 4; ++i) {
        const int tt = wave + 4 * i;        // 0..15
        const int qx = tt & 1;
        const int ry = (tt >> 1) & 1;
        const int fx = (tt >> 2) & 1;
        const int fy = tt >> 3;
        v8f acc = {0.f, 0.f, 0.f, 0.f, 0.f, 0.f, 0.f, 0.f};
#pragma unroll
        for (int k = 0; k < 12; ++k) {
            const int cf = 4 * k + khalf;       // filter-table column
            const int cr = 32 * ry + cf;        // intermediate row
            v2f a;  // A(16x4): filter matrix (same layout expr as row-pass B)
            a.x = s_flt[fy][mrow][cf];
            a.y = s_flt[fy][mrow][cf + 1];
            v2f b;  // B(4x16): intermediate data
            b.x = s_int[fx][cr][qx * 16 + mrow];
            b.y = s_int[fx][cr + 1][qx * 16 + mrow];
            acc = __builtin_amdgcn_wmma_f32_16x16x4_f32(
                false, a, false, b, (short)0, acc, false, false);
        }
        // Destination subband: aa=(lo,lo), da=(hiY,loX), ad=(loY,hiX), dd=(hi,hi)
        float* dst;
        if (fy == 0 && fx == 0) {
            dst = aa + (size_t)img * h * w;
        } else {
            const int sb = (fx == 0) ? 0 : (fy == 0 ? 1 : 2);  // da, ad, dd
            dst = hi_base + (size_t)sb * slab + (size_t)img * h * w;
        }
#pragma unroll
        for (int v = 0; v < 8; ++v) {
            const int oy = oy0 + ry * 16 + dhalf + v;
            const int ox = ox0 + qx * 16 + mrow;
            dst[(size_t)oy * w + ox] = acc[v];
        }
    }
}

extern "C" void kernel_launch(void* const* d_in, const int* in_sizes, int n_in,
                              void* d_out, int out_size, void* d_ws, size_t ws_size,
                              hipStream_t stream) {
    (void)in_sizes; (void)n_in; (void)out_size; (void)ws_size;
    const float* x = (const float*)d_in[0];   // (8,3,1024,1024) -> 24 images
    float* out = (float*)d_out;
    float* ws  = (float*)d_ws;

    const int NI = 24;
    // d_out layout (flat, return order): a | h3[3][24][128][128] | h2[...256] | h1[...512]
    float* a_out = out;                              // 24*128*128 = 393216
    float* h3    = out + 393216;                     // 3 slabs of 393216
    float* h2    = out + 1572864;                    // 3 slabs of 1572864
    float* h1    = out + 6291456;                    // 3 slabs of 6291456

    float* aa1 = ws;                                 // 24*512*512 floats
    float* aa2 = ws + (size_t)NI * 512 * 512;        // 24*256*256 floats

    dim3 blk(128, 1, 1);
    // Level 1: (24,1024,1024) -> aa1 + h1
    dwt2_level_kernel<<<dim3(512 / OT, 512 / OT, NI), blk, 0, stream>>>(
        x, aa1, h1, (size_t)NI * 512 * 512, 1024, 1024);
    // Level 2: aa1 -> aa2 + h2
    dwt2_level_kernel<<<dim3(256 / OT, 256 / OT, NI), blk, 0, stream>>>(
        aa1, aa2, h2, (size_t)NI * 256 * 256, 512, 512);
    // Level 3: aa2 -> a_out + h3
    dwt2_level_kernel<<<dim3(128 / OT, 128 / OT, NI), blk, 0, stream>>>(
        aa2, a_out, h3, (size_t)NI * 128 * 128, 256, 256);
}